// ZeroQueryAndGroup_23922967839002
// MI455X (gfx1250) — compile-verified
//
#include <hip/hip_runtime.h>

// Problem constants (from reference setup_inputs)
#define B_  4
#define N_  16384
#define P_  2048
#define C_  32
#define NS  32
#define R2  0.04f   // RADIUS^2, RADIUS = 0.2

typedef __attribute__((ext_vector_type(2))) float v2f;
typedef __attribute__((ext_vector_type(8))) float v8f;

// ---------------------------------------------------------------------------
// Kernel 1: ball query via V_WMMA_F32_16X16X4_F32.
// One wave (32 threads) handles 16 query points; loops over N in 32-point
// chunks (two WMMAs per iteration). Each WMMA computes
//   D[m][n] = -2 q_m . x_n + |x_n|^2
// for a 16x16 tile; in-ball iff D < r^2 - |q_m|^2. Lane m (<16) owns query m
// and appends ascending in-ball padded indices (j+1) until NS found.
// Early exit when every query in the wave is full.
// ---------------------------------------------------------------------------
__global__ __launch_bounds__(32)
void ZeroQueryAndGroup_ballquery_wmma(const float* __restrict__ xyz,
                                      const float* __restrict__ new_xyz,
                                      int* __restrict__ idx_ws,
                                      float* __restrict__ valid_out)
{
    const int  lane  = threadIdx.x;        // 0..31
    const int  b     = blockIdx.y;
    const int  qTile = blockIdx.x;         // 0..P/16-1
    const int  m     = lane & 15;
    const bool hi    = lane >= 16;
    const int  q     = qTile * 16 + m;

    // --- A matrix: row m = [qx, qy, qz, 1] ---------------------------------
    const float qx = new_xyz[(b * P_ + q) * 3 + 0];
    const float qy = new_xyz[(b * P_ + q) * 3 + 1];
    const float qz = new_xyz[(b * P_ + q) * 3 + 2];
    v2f a;
    a.x = hi ? qz   : qx;   // VGPR0: K0 (lanes 0-15) / K2 (lanes 16-31)
    a.y = hi ? 1.0f : qy;   // VGPR1: K1 / K3

    // Per-query threshold; broadcast to per-D-row thresholds.
    // D-row VGPR v holds query v (lanes 0-15) and query v+8 (lanes 16-31).
    const float thresh = R2 - (qx * qx + qy * qy + qz * qz);
    float thr[8];
#pragma unroll
    for (int v = 0; v < 8; ++v)
        thr[v] = __shfl(thresh, v + (hi ? 8 : 0), 32);

    unsigned cnt   = hi ? (unsigned)NS : 0u;  // lanes 16-31: permanently "full"
    int      first = 0;
    const int base = (b * P_ + q) * NS;

    const float* xb = xyz + (size_t)b * N_ * 3;

    for (int chunk = 0; chunk < N_; chunk += 32) {
        // --- B matrices for points chunk..chunk+15 and chunk+16..chunk+31 --
        // col n = [-2x, -2y, -2z, |x|^2]; both halves load the same 16 pts.
        const int   j0  = chunk + m;
        const int   j1  = chunk + 16 + m;
        const float p0x = xb[j0 * 3 + 0];
        const float p0y = xb[j0 * 3 + 1];
        const float p0z = xb[j0 * 3 + 2];
        const float p1x = xb[j1 * 3 + 0];
        const float p1y = xb[j1 * 3 + 1];
        const float p1z = xb[j1 * 3 + 2];
        const float sq0 = p0x * p0x + p0y * p0y + p0z * p0z;
        const float sq1 = p1x * p1x + p1y * p1y + p1z * p1z;
        v2f bm0, bm1;
        bm0.x = hi ? (-2.0f * p0z) : (-2.0f * p0x);  // VGPR0: row0 / row2
        bm0.y = hi ? sq0           : (-2.0f * p0y);  // VGPR1: row1 / row3
        bm1.x = hi ? (-2.0f * p1z) : (-2.0f * p1x);
        bm1.y = hi ? sq1           : (-2.0f * p1y);

        if (chunk + 1024 < N_)
            __builtin_prefetch(&xb[(chunk + 1024 + m) * 3], 0, 3);

        v8f c0 = {}, c1 = {};
        c0 = __builtin_amdgcn_wmma_f32_16x16x4_f32(false, a, false, bm0,
                                                   (short)0, c0, false, false);
        c1 = __builtin_amdgcn_wmma_f32_16x16x4_f32(false, a, false, bm1,
                                                   (short)0, c1, false, false);

        // One ballot per D-row VGPR (wave-uniform 32-bit masks).
        const unsigned a0 = __builtin_amdgcn_ballot_w32(c0[0] < thr[0]);
        const unsigned a1 = __builtin_amdgcn_ballot_w32(c0[1] < thr[1]);
        const unsigned a2 = __builtin_amdgcn_ballot_w32(c0[2] < thr[2]);
        const unsigned a3 = __builtin_amdgcn_ballot_w32(c0[3] < thr[3]);
        const unsigned a4 = __builtin_amdgcn_ballot_w32(c0[4] < thr[4]);
        const unsigned a5 = __builtin_amdgcn_ballot_w32(c0[5] < thr[5]);
        const unsigned a6 = __builtin_amdgcn_ballot_w32(c0[6] < thr[6]);
        const unsigned a7 = __builtin_amdgcn_ballot_w32(c0[7] < thr[7]);
        const unsigned b0 = __builtin_amdgcn_ballot_w32(c1[0] < thr[0]);
        const unsigned b1 = __builtin_amdgcn_ballot_w32(c1[1] < thr[1]);
        const unsigned b2 = __builtin_amdgcn_ballot_w32(c1[2] < thr[2]);
        const unsigned b3 = __builtin_amdgcn_ballot_w32(c1[3] < thr[3]);
        const unsigned b4 = __builtin_amdgcn_ballot_w32(c1[4] < thr[4]);
        const unsigned b5 = __builtin_amdgcn_ballot_w32(c1[5] < thr[5]);
        const unsigned b6 = __builtin_amdgcn_ballot_w32(c1[6] < thr[6]);
        const unsigned b7 = __builtin_amdgcn_ballot_w32(c1[7] < thr[7]);

        // Lane m picks row (m&7), half (m>>3): 7-select tree + shift, twice.
        const unsigned sh = (lane & 8) ? 16 : 0;
        unsigned ta0 = (lane & 1) ? a1 : a0;
        unsigned ta1 = (lane & 1) ? a3 : a2;
        unsigned ta2 = (lane & 1) ? a5 : a4;
        unsigned ta3 = (lane & 1) ? a7 : a6;
        unsigned ua0 = (lane & 2) ? ta1 : ta0;
        unsigned ua1 = (lane & 2) ? ta3 : ta2;
        const unsigned lo16 = (((lane & 4) ? ua1 : ua0) >> sh) & 0xFFFFu;
        unsigned tb0 = (lane & 1) ? b1 : b0;
        unsigned tb1 = (lane & 1) ? b3 : b2;
        unsigned tb2 = (lane & 1) ? b5 : b4;
        unsigned tb3 = (lane & 1) ? b7 : b6;
        unsigned ub0 = (lane & 2) ? tb1 : tb0;
        unsigned ub1 = (lane & 2) ? tb3 : tb2;
        const unsigned hi16 = (((lane & 4) ? ub1 : ub0) >> sh) & 0xFFFFu;

        unsigned mask32 = lo16 | (hi16 << 16);   // bit k -> point chunk + k

        // Append ascending in-ball padded indices (j+1). Divergent but short.
        while (mask32 && cnt < NS) {
            const unsigned bit = __builtin_ctz(mask32);
            mask32 &= (mask32 - 1);
            const int pj = chunk + (int)bit + 1;   // +1: padded index space
            if (cnt == 0) first = pj;
            idx_ws[base + (int)cnt] = pj;
            ++cnt;
        }

        // Uniform early exit once all 16 queries in the wave are full.
        if (__builtin_amdgcn_ballot_w32(cnt < NS) == 0) break;
    }

    if (!hi) {
        // Fill remaining slots with first index (0 if no neighbor found).
#pragma unroll 1
        for (unsigned s = cnt; s < NS; ++s) idx_ws[base + (int)s] = first;
        valid_out[b * P_ + q] = (cnt > 0u) ? 1.0f : 0.0f;
    }
}

// ---------------------------------------------------------------------------
// Kernel 2: gather/group. Pure bandwidth; coalesced on s (innermost).
// out[b][ch][p][s]: ch 0-2 = xyz_pad[idx]-new_xyz (zero if idx==0),
//                   ch 3-34 = features[ch-3][idx-1] (zero if idx==0).
// ---------------------------------------------------------------------------
__global__ __launch_bounds__(256)
void ZeroQueryAndGroup_gather(const float* __restrict__ xyz,
                              const float* __restrict__ new_xyz,
                              const float* __restrict__ feat,
                              const int* __restrict__ idx_ws,
                              float* __restrict__ out)
{
    const int t  = blockIdx.x * 256 + (int)threadIdx.x;  // over P*NS
    const int s  = t & (NS - 1);
    const int p  = t >> 5;
    const int ch = blockIdx.y;                           // 0..34
    const int b  = blockIdx.z;

    const int idx = idx_ws[(b * P_ + p) * NS + s];
    float val = 0.0f;
    if (idx != 0) {
        const int j = idx - 1;
        if (ch < 3) {
            val = xyz[((size_t)b * N_ + j) * 3 + ch]
                - new_xyz[(b * P_ + p) * 3 + ch];
        } else {
            const int cc = ch - 3;
            val = feat[((size_t)(b * C_ + cc)) * N_ + j];
        }
    }
    out[(((size_t)b * 35 + ch) * P_ + p) * NS + s] = val;
}

// ---------------------------------------------------------------------------
extern "C" void kernel_launch(void* const* d_in, const int* in_sizes, int n_in,
                              void* d_out, int out_size, void* d_ws, size_t ws_size,
                              hipStream_t stream)
{
    const float* xyz     = (const float*)d_in[0];  // (B, N, 3)
    const float* new_xyz = (const float*)d_in[1];  // (B, P, 3)
    const float* feat    = (const float*)d_in[2];  // (B, C, N)

    float* out       = (float*)d_out;                        // (B, 35, P, NS)
    float* valid_out = out + (size_t)B_ * 35 * P_ * NS;      // (B, P) as 0/1
    int*   idx_ws    = (int*)d_ws;                           // (B, P, NS) ints

    dim3 g1(P_ / 16, B_);
    ZeroQueryAndGroup_ballquery_wmma<<<g1, 32, 0, stream>>>(
        xyz, new_xyz, idx_ws, valid_out);

    dim3 g2((P_ * NS) / 256, 35, B_);
    ZeroQueryAndGroup_gather<<<g2, 256, 0, stream>>>(
        xyz, new_xyz, feat, idx_ws, out);
}